// MACCell_52321291600663
// MI455X (gfx1250) — compile-verified
//
#include <hip/hip_runtime.h>
#include <hip/hip_bf16.h>

// MAC cell, algebraically collapsed: the read-attention logits reduce to a
// per-batch GEMV  r_logits[b,k] = kb[b,k]·w[b] + const[b], so the two huge
// [B,K,D] GEMMs of the reference are replaced by six tiny f32 WMMA GEMMs and
// two memory-bound streaming passes over kb (268 MB -> ~12-23us at 23.3 TB/s).

#define Bb 128
#define Dd 512
#define Ss 64
#define Kk 1024

typedef __attribute__((ext_vector_type(2))) float v2f;
typedef __attribute__((ext_vector_type(8))) float v8f;

// ---------------------------------------------------------------------------
// Generic small GEMM: C[M,N] = A[M,Kdim] x B (+bias1 +bias2[step])
// One wave computes one 16x16 C tile via V_WMMA_F32_16X16X4_F32.
// K-loop unrolled 8x (32 k-values / iter): all loads issued, then 8 WMMAs.
// A layout (16x4 f32): lanes 0-15 row=l cols k0,k0+1 ; lanes 16-31 cols k0+2,k0+3
// B layout (4x16 f32): lanes 0-15 rows k0,k0+1 col=n0+l ; lanes 16-31 rows k0+2,k0+3
// C layout: VGPR i -> row m0 + 8*half + i, col n0 + l
// TRANSB==1: B stored [N,Kdim] row-major, access B[n*ldb+k].
// ---------------------------------------------------------------------------
template <int TRANSB>
__global__ void wmma_gemm_f32(const float* __restrict__ A, const float* __restrict__ Bm,
                              float* __restrict__ C,
                              const float* __restrict__ bias1,
                              const float* __restrict__ bias2_base,
                              const int* __restrict__ step_ptr, int bias2_stride,
                              int M, int N, int Kdim, int lda, int ldb, int ldc)
{
    const int wave  = (blockIdx.x * blockDim.x + threadIdx.x) >> 5;
    const int lane  = threadIdx.x & 31;
    const int tilesN = N >> 4;
    const int m0 = (wave / tilesN) << 4;
    const int n0 = (wave % tilesN) << 4;
    if (m0 >= M) return;                       // wave-uniform: EXEC stays full
    const int half = lane >> 4;                // 0 or 1
    const int l    = lane & 15;

    const float* bias2 = nullptr;
    if (bias2_base) bias2 = bias2_base + (size_t)(*step_ptr) * bias2_stride;

    // Per-lane base pointers; inner loop is pure pointer arithmetic.
    const float* aBase = A + (size_t)(m0 + l) * lda + 2 * half;
    const float* bBase = TRANSB ? (Bm + (size_t)(n0 + l) * ldb + 2 * half)
                                : (Bm + (size_t)(2 * half) * ldb + (n0 + l));

    v8f acc = {};
    for (int k0 = 0; k0 < Kdim; k0 += 32) {    // 8 WMMA steps per iteration
        v2f a[8], b[8];
#pragma unroll
        for (int t = 0; t < 8; ++t) {
            const int k = k0 + 4 * t;
            a[t].x = aBase[k];
            a[t].y = aBase[k + 1];
            if (TRANSB) {
                b[t].x = bBase[k];
                b[t].y = bBase[k + 1];
            } else {
                const float* bp = bBase + (size_t)k * ldb;
                b[t].x = bp[0];
                b[t].y = bp[ldb];
            }
        }
#pragma unroll
        for (int t = 0; t < 8; ++t) {
            acc = __builtin_amdgcn_wmma_f32_16x16x4_f32(
                      false, a[t], false, b[t], (short)0, acc, false, false);
        }
    }
#pragma unroll
    for (int i = 0; i < 8; ++i) {
        int row = m0 + 8 * half + i;
        int col = n0 + l;
        float v = acc[i];
        if (bias1) v += bias1[col];
        if (bias2) v += bias2[col];
        C[(size_t)row * ldc + col] = v;
    }
}

// dst[r, 0:n1]=a[r], dst[r, n1:n1+n2]=b[r]
__global__ void pack_pair_kernel(float* __restrict__ dst,
                                 const float* __restrict__ a, const float* __restrict__ b,
                                 int rows, int n1, int n2)
{
    int idx = blockIdx.x * blockDim.x + threadIdx.x;
    int w = n1 + n2;
    if (idx >= rows * w) return;
    int r = idx / w, c = idx - r * w;
    dst[idx] = (c < n1) ? a[(size_t)r * n1 + c] : b[(size_t)r * n2 + (c - n1)];
}

// ---------------------------------------------------------------------------
// ControlUnit tail: logits over context, softmax(S=64), new_control, and
// g = new_control * W_rattn (feeds the collapsed ReadUnit). One block / batch.
// ---------------------------------------------------------------------------
__global__ void control_unit_kernel(const float* __restrict__ cq_proj,
                                    const float* __restrict__ W_cattn,
                                    const float* __restrict__ b_cattn,
                                    const float* __restrict__ context,
                                    const unsigned char* __restrict__ q_mask,
                                    const float* __restrict__ W_rattn,
                                    float* __restrict__ new_control,
                                    float* __restrict__ g)
{
    __shared__ float h[Dd];
    __shared__ float lg[Ss];
    __shared__ float cw[Ss];
    const int b = blockIdx.x;
    const int tid = threadIdx.x;              // 256 threads
    h[tid]       = cq_proj[(size_t)b * Dd + tid] * W_cattn[tid];
    h[tid + 256] = cq_proj[(size_t)b * Dd + tid + 256] * W_cattn[tid + 256];
    __syncthreads();

    const int wave = tid >> 5, lane = tid & 31;
    const float* ctx = context + (size_t)b * Ss * Dd;
    for (int s = wave; s < Ss; s += 8) {
        const float* row = ctx + (size_t)s * Dd;
        float p = 0.f;
#pragma unroll
        for (int j = lane; j < Dd; j += 32) p += row[j] * h[j];
        for (int off = 16; off > 0; off >>= 1) p += __shfl_xor(p, off, 32);
        if (lane == 0)
            lg[s] = q_mask[(size_t)b * Ss + s] ? (p + b_cattn[0]) : -10000.0f;
    }
    __syncthreads();
    if (tid == 0) {                            // S=64, trivial serial softmax
        float m = lg[0];
        for (int s = 1; s < Ss; ++s) m = fmaxf(m, lg[s]);
        float sum = 0.f;
        for (int s = 0; s < Ss; ++s) { float e = __expf(lg[s] - m); cw[s] = e; sum += e; }
        float inv = 1.0f / sum;
        for (int s = 0; s < Ss; ++s) cw[s] *= inv;
    }
    __syncthreads();
    for (int d = tid; d < Dd; d += 256) {
        float acc = 0.f;
#pragma unroll 8
        for (int s = 0; s < Ss; ++s) acc += cw[s] * ctx[(size_t)s * Dd + d];
        new_control[(size_t)b * Dd + d] = acc;
        g[(size_t)b * Dd + d] = acc * W_rattn[d];
    }
}

// u = mem_proj*A1 + A2 ; const[b] = b_kb·u + b_cat·g + b_rattn
__global__ void u_const_kernel(const float* __restrict__ mem_proj,
                               const float* __restrict__ A1, const float* __restrict__ A2,
                               const float* __restrict__ g,
                               const float* __restrict__ b_kb, const float* __restrict__ b_cat,
                               const float* __restrict__ b_rattn,
                               float* __restrict__ u, float* __restrict__ constv)
{
    __shared__ float red[256];
    const int b = blockIdx.x, tid = threadIdx.x;
    float p = 0.f;
    for (int e = tid; e < Dd; e += 256) {
        size_t i = (size_t)b * Dd + e;
        float ue = mem_proj[i] * A1[i] + A2[i];
        u[i] = ue;
        p += b_kb[e] * ue + b_cat[e] * g[i];
    }
    red[tid] = p; __syncthreads();
    for (int s = 128; s > 0; s >>= 1) { if (tid < s) red[tid] += red[tid + s]; __syncthreads(); }
    if (tid == 0) constv[b] = red[0] + b_rattn[0];
}

// ---------------------------------------------------------------------------
// Pass 1 over kb: logits = kb[b] @ w[b] + const[b], softmax over K=1024.
// One block (512 threads = 16 waves) per batch; coalesced 128B/wave row reads.
// ---------------------------------------------------------------------------
__global__ void kb_attn_kernel(const float* __restrict__ kb, const float* __restrict__ wv,
                               const float* __restrict__ constv,
                               const unsigned char* __restrict__ kb_mask,
                               float* __restrict__ attn)
{
    __shared__ float ws_[Dd];
    __shared__ float lg[Kk];
    __shared__ float red[512];
    const int b = blockIdx.x, tid = threadIdx.x;   // 512 threads
    ws_[tid] = wv[(size_t)b * Dd + tid];
    __syncthreads();

    const int wave = tid >> 5, lane = tid & 31;
    const float* kbb = kb + (size_t)b * Kk * Dd;
    const float cb = constv[b];
    for (int k = wave; k < Kk; k += 16) {
        const float* row = kbb + (size_t)k * Dd;
        __builtin_prefetch(row + 16 * Dd, 0, 1);    // next row for this wave
        float p = 0.f;
#pragma unroll
        for (int j = lane; j < Dd; j += 32) p += row[j] * ws_[j];
        for (int off = 16; off > 0; off >>= 1) p += __shfl_xor(p, off, 32);
        if (lane == 0)
            lg[k] = kb_mask[(size_t)b * Kk + k] ? (p + cb) : -10000.0f;
    }
    __syncthreads();

    float l1 = lg[tid], l2 = lg[tid + 512];
    red[tid] = fmaxf(l1, l2); __syncthreads();
    for (int s = 256; s > 0; s >>= 1) { if (tid < s) red[tid] = fmaxf(red[tid], red[tid + s]); __syncthreads(); }
    float m = red[0]; __syncthreads();
    float e1 = __expf(l1 - m), e2 = __expf(l2 - m);
    red[tid] = e1 + e2; __syncthreads();
    for (int s = 256; s > 0; s >>= 1) { if (tid < s) red[tid] += red[tid + s]; __syncthreads(); }
    float inv = 1.0f / red[0];
    attn[(size_t)b * Kk + tid]       = e1 * inv;
    attn[(size_t)b * Kk + tid + 512] = e2 * inv;
}

// Pass 2 over kb: read_vector[b,d] = sum_k attn[b,k]*kb[b,k,d] (L2-resident slice)
__global__ void read_vec_kernel(const float* __restrict__ kb,
                                const float* __restrict__ attn,
                                float* __restrict__ rv)
{
    __shared__ float aw[Kk];
    const int b = blockIdx.x, tid = threadIdx.x;   // 512 threads, one d each
    aw[tid]       = attn[(size_t)b * Kk + tid];
    aw[tid + 512] = attn[(size_t)b * Kk + tid + 512];
    __syncthreads();
    const float* kbb = kb + (size_t)b * Kk * Dd;
    float acc = 0.f;
#pragma unroll 8
    for (int k = 0; k < Kk; ++k) acc += aw[k] * kbb[(size_t)k * Dd + tid];
    rv[(size_t)b * Dd + tid] = acc;
}

// ---------------------------------------------------------------------------
static inline void launch_gemm(const float* A, const float* Bm, float* C,
                               const float* b1, const float* b2b, const int* stepp, int b2s,
                               int M, int N, int Kdim, int lda, int ldb, int ldc,
                               int transB, hipStream_t stream)
{
    int tiles = (M / 16) * (N / 16);
    int wpb = 8;                                   // 8 waves / block (256 thr)
    int blocks = (tiles + wpb - 1) / wpb;
    if (transB)
        wmma_gemm_f32<1><<<blocks, wpb * 32, 0, stream>>>(A, Bm, C, b1, b2b, stepp, b2s,
                                                          M, N, Kdim, lda, ldb, ldc);
    else
        wmma_gemm_f32<0><<<blocks, wpb * 32, 0, stream>>>(A, Bm, C, b1, b2b, stepp, b2s,
                                                          M, N, Kdim, lda, ldb, ldc);
}

extern "C" void kernel_launch(void* const* d_in, const int* in_sizes, int n_in,
                              void* d_out, int out_size, void* d_ws, size_t ws_size,
                              hipStream_t stream) {
    (void)in_sizes; (void)n_in; (void)out_size; (void)ws_size;
    const int*   step     = (const int*)  d_in[0];
    const float* context  = (const float*)d_in[1];
    const float* question = (const float*)d_in[2];
    const float* kb       = (const float*)d_in[3];
    const float* control  = (const float*)d_in[4];
    const float* memory   = (const float*)d_in[5];
    const unsigned char* kb_mask = (const unsigned char*)d_in[6];
    const unsigned char* q_mask  = (const unsigned char*)d_in[7];
    const float* W_cq    = (const float*)d_in[8];
    const float* b_cq    = (const float*)d_in[9];
    const float* step_emb= (const float*)d_in[10];
    const float* W_cattn = (const float*)d_in[11];
    const float* b_cattn = (const float*)d_in[12];
    const float* W_mem   = (const float*)d_in[13];
    const float* b_mem   = (const float*)d_in[14];
    const float* W_kb    = (const float*)d_in[15];
    const float* b_kb    = (const float*)d_in[16];
    const float* W_cat   = (const float*)d_in[17];
    const float* b_cat   = (const float*)d_in[18];
    const float* W_rattn = (const float*)d_in[19];
    const float* b_rattn = (const float*)d_in[20];
    const float* W_wr    = (const float*)d_in[21];
    const float* b_wr    = (const float*)d_in[22];

    float* out_nc  = (float*)d_out;                 // new_control [B,D]
    float* out_nm  = out_nc + Bb * Dd;              // new_memory  [B,D]
    float* out_att = out_nc + 2 * Bb * Dd;          // attn        [B,K]

    float* ws       = (float*)d_ws;
    float* cq_pack  = ws;                   // B*2D = 131072
    float* cq_proj  = cq_pack + Bb * 2 * Dd;
    float* mem_proj = cq_proj + Bb * Dd;
    float* A1       = mem_proj + Bb * Dd;
    float* A2       = A1 + Bb * Dd;
    float* u        = A2 + Bb * Dd;
    float* wv       = u + Bb * Dd;
    float* gv       = wv + Bb * Dd;
    float* rv       = gv + Bb * Dd;
    float* constv   = rv + Bb * Dd;         // B floats
    float* mw_pack  = constv + Bb;          // B*2D

    // 1) cq = [control | question]
    {
        int n = Bb * 2 * Dd;
        pack_pair_kernel<<<(n + 255) / 256, 256, 0, stream>>>(cq_pack, control, question, Bb, Dd, Dd);
    }
    // 2) cq_proj = cq @ W_cq + b_cq + step_emb[step]
    launch_gemm(cq_pack, W_cq, cq_proj, b_cq, step_emb, step, Dd,
                Bb, Dd, 2 * Dd, 2 * Dd, Dd, Dd, 0, stream);
    // 3) control attention -> new_control (d_out), g = nc * W_rattn
    control_unit_kernel<<<Bb, 256, 0, stream>>>(cq_proj, W_cattn, b_cattn, context,
                                                q_mask, W_rattn, out_nc, gv);
    // 4) mem_proj = memory @ W_mem + b_mem
    launch_gemm(memory, W_mem, mem_proj, b_mem, nullptr, nullptr, 0,
                Bb, Dd, Dd, Dd, Dd, Dd, 0, stream);
    // 5) A1 = g @ W_cat[:D]^T ; A2 = g @ W_cat[D:]^T
    launch_gemm(gv, W_cat, A1, nullptr, nullptr, nullptr, 0,
                Bb, Dd, Dd, Dd, Dd, Dd, 1, stream);
    launch_gemm(gv, W_cat + (size_t)Dd * Dd, A2, nullptr, nullptr, nullptr, 0,
                Bb, Dd, Dd, Dd, Dd, Dd, 1, stream);
    // 6) u = mem_proj*A1 + A2 ; const[b]
    u_const_kernel<<<Bb, 256, 0, stream>>>(mem_proj, A1, A2, gv, b_kb, b_cat, b_rattn,
                                           u, constv);
    // 7) w = u @ W_kb^T
    launch_gemm(u, W_kb, wv, nullptr, nullptr, nullptr, 0,
                Bb, Dd, Dd, Dd, Dd, Dd, 1, stream);
    // 8) logits + softmax over kb  (streams 268 MB — the roofline term)
    kb_attn_kernel<<<Bb, 512, 0, stream>>>(kb, wv, constv, kb_mask, out_att);
    // 9) read_vector = attn @ kb
    read_vec_kernel<<<Bb, 512, 0, stream>>>(kb, out_att, rv);
    // 10) new_memory = [memory | read_vector] @ W_wr + b_wr
    {
        int n = Bb * 2 * Dd;
        pack_pair_kernel<<<(n + 255) / 256, 256, 0, stream>>>(mw_pack, memory, rv, Bb, Dd, Dd);
    }
    launch_gemm(mw_pack, W_wr, out_nm, b_wr, nullptr, nullptr, 0,
                Bb, Dd, 2 * Dd, 2 * Dd, Dd, Dd, 0, stream);
}